// DistMult_1743756722750
// MI455X (gfx1250) — compile-verified
//
#include <hip/hip_runtime.h>
#include <hip/hip_bf16.h>

// DistMult scoring on gfx1250:
//   score_e = src_e^T * W[rel_e] * dst_e   (DIM = 64, 64 relations)
// Strategy: bucket edges by relation (hist -> scan -> scatter), then score
// 16-edge tiles per relation as a real GEMM using V_WMMA_F32_16X16X4_F32:
//   T(16x64) = Dst(16x64) x W_r^T(64x64);  score_e = dot(Src_e, T_e).

typedef __attribute__((ext_vector_type(2))) float v2f;
typedef __attribute__((ext_vector_type(8))) float v8f;

#define DIM 64
#define NRELS 64
#define LDS_STRIDE 68          // 64 + 4 pad: kills stride-64 bank conflicts
#define WAVES_PER_BLOCK 4
#define TILE_M 16

__global__ void zero_ws_kernel(int* counts, int* fill) {
  int i = threadIdx.x;
  if (i < NRELS) { counts[i] = 0; fill[i] = 0; }
}

__global__ void hist_kernel(const int* __restrict__ trip, int* __restrict__ counts, int E) {
  int i = blockIdx.x * blockDim.x + threadIdx.x;
  if (i < E) atomicAdd(&counts[trip[3 * i + 1]], 1);
}

__global__ void scan_kernel(const int* __restrict__ counts, int* __restrict__ offsets) {
  if (threadIdx.x == 0 && blockIdx.x == 0) {
    int acc = 0;
    for (int r = 0; r < NRELS; ++r) { offsets[r] = acc; acc += counts[r]; }
    offsets[NRELS] = acc;
  }
}

__global__ void scatter_kernel(const int* __restrict__ trip, const int* __restrict__ offsets,
                               int* __restrict__ fill, int* __restrict__ perm, int E) {
  int i = blockIdx.x * blockDim.x + threadIdx.x;
  if (i < E) {
    int r = trip[3 * i + 1];
    int pos = offsets[r] + atomicAdd(&fill[r], 1);
    perm[pos] = i;          // slot order nondeterministic, score per edge is not
  }
}

__launch_bounds__(WAVES_PER_BLOCK * 32)
__global__ void score_kernel(const int* __restrict__ trip,
                             const float* __restrict__ emb,
                             const float* __restrict__ W,
                             const int* __restrict__ offsets,
                             const int* __restrict__ perm,
                             float* __restrict__ out) {
  __shared__ float ldsW[DIM * LDS_STRIDE];                       // 17408 B
  __shared__ float ldsT[WAVES_PER_BLOCK * TILE_M * LDS_STRIDE];  // 17408 B

  const int rel   = blockIdx.x;
  const int begin = offsets[rel];
  const int n     = offsets[rel + 1] - begin;

  // ---- cooperative load of W_r (64x64 f32) into padded LDS ----
  {
    const float* Wr = W + (size_t)rel * DIM * DIM;
    int row  = threadIdx.x >> 1;       // 2 threads per row
    int half = threadIdx.x & 1;
    #pragma unroll
    for (int i = 0; i < 8; ++i) {
      int k = half * 32 + i * 4;
      float4 v = *(const float4*)(Wr + row * DIM + k);
      *(float4*)(&ldsW[row * LDS_STRIDE + k]) = v;
    }
  }
  __syncthreads();

  const int wave = threadIdx.x >> 5;
  const int lane = threadIdx.x & 31;
  const int m    = lane & 15;          // row/col index within 16
  const int hi   = lane >> 4;          // lane half select
  float* myT = &ldsT[wave * TILE_M * LDS_STRIDE];

  const int ntiles = (n + TILE_M - 1) >> 4;
  for (int t = blockIdx.y * WAVES_PER_BLOCK + wave; t < ntiles;
       t += gridDim.y * WAVES_PER_BLOCK) {

    // ---- gather 16 dst rows into myT (2 lanes per row, 32 floats each) ----
    {
      int r = lane >> 1;
      int h = lane & 1;
      int slot = t * TILE_M + r;
      int eid = (slot < n) ? perm[begin + slot] : 0;   // clamp tail to edge 0
      int dnode = trip[3 * eid + 2];
      const float* drow = emb + (size_t)dnode * DIM;
      #pragma unroll
      for (int i = 0; i < 8; ++i) {
        int k = h * 32 + i * 4;
        float4 v = *(const float4*)(drow + k);
        *(float4*)(&myT[r * LDS_STRIDE + k]) = v;
      }
    }
    asm volatile("s_wait_dscnt 0" ::: "memory");  // LDS writes visible wave-wide

    // ---- T = Dst(16x64) x W_r^T : 16 K-steps x 4 N-tiles of WMMA f32 ----
    // A (16x4): lanes0-15 hold (M=m, K=kk..kk+1), lanes16-31 (M=m, K=kk+2..3)
    // B (4x16): lanes0-15 hold (K=kk..kk+1, N=m), lanes16-31 (K=kk+2..3, N=m)
    // B[k][n] = W_r[n][k]  (so T[e][d] = sum_k Dst[e][k] * W_r[d][k])
    v8f acc0 = {}; v8f acc1 = {}; v8f acc2 = {}; v8f acc3 = {};
    #pragma unroll
    for (int kk = 0; kk < DIM; kk += 4) {
      int kb = kk + 2 * hi;
      v2f a  = *(const v2f*)(&myT[m * LDS_STRIDE + kb]);
      v2f b0 = *(const v2f*)(&ldsW[( 0 + m) * LDS_STRIDE + kb]);
      v2f b1 = *(const v2f*)(&ldsW[(16 + m) * LDS_STRIDE + kb]);
      v2f b2 = *(const v2f*)(&ldsW[(32 + m) * LDS_STRIDE + kb]);
      v2f b3 = *(const v2f*)(&ldsW[(48 + m) * LDS_STRIDE + kb]);
      acc0 = __builtin_amdgcn_wmma_f32_16x16x4_f32(false, a, false, b0, (short)0, acc0, false, false);
      acc1 = __builtin_amdgcn_wmma_f32_16x16x4_f32(false, a, false, b1, (short)0, acc1, false, false);
      acc2 = __builtin_amdgcn_wmma_f32_16x16x4_f32(false, a, false, b2, (short)0, acc2, false, false);
      acc3 = __builtin_amdgcn_wmma_f32_16x16x4_f32(false, a, false, b3, (short)0, acc3, false, false);
    }

    // ---- spill T back into myT (C/D layout: VGPR j -> M=j(+8), N=lane%16) ----
    asm volatile("s_wait_dscnt 0" ::: "memory");
    #pragma unroll
    for (int j = 0; j < 8; ++j) {
      int mm = j + 8 * hi;
      myT[mm * LDS_STRIDE +  0 + m] = acc0[j];
      myT[mm * LDS_STRIDE + 16 + m] = acc1[j];
      myT[mm * LDS_STRIDE + 32 + m] = acc2[j];
      myT[mm * LDS_STRIDE + 48 + m] = acc3[j];
    }
    asm volatile("s_wait_dscnt 0" ::: "memory");

    // ---- score_e = dot(src_row_e, T_row_e), one edge per low lane ----
    if (hi == 0) {
      int slot = t * TILE_M + m;
      if (slot < n) {
        int eid = perm[begin + slot];
        int snode = trip[3 * eid + 0];
        const float* srow = emb + (size_t)snode * DIM;
        float s = 0.f;
        #pragma unroll
        for (int d = 0; d < DIM; d += 4) {
          float4 sv = *(const float4*)(srow + d);
          s += sv.x * myT[m * LDS_STRIDE + d + 0];
          s += sv.y * myT[m * LDS_STRIDE + d + 1];
          s += sv.z * myT[m * LDS_STRIDE + d + 2];
          s += sv.w * myT[m * LDS_STRIDE + d + 3];
        }
        out[eid] = s;
      }
    }
    asm volatile("s_wait_dscnt 0" ::: "memory");  // before next tile overwrites myT
  }
}

extern "C" void kernel_launch(void* const* d_in, const int* in_sizes, int n_in,
                              void* d_out, int out_size, void* d_ws, size_t ws_size,
                              hipStream_t stream) {
  const int*   trip = (const int*)d_in[0];   // (E,3) int32: [src, rel, dst]
  const float* emb  = (const float*)d_in[1]; // (NUM_NODES, 64) f32
  const float* W    = (const float*)d_in[2]; // (64, 64, 64) f32
  float* out = (float*)d_out;                // (E,) f32
  const int E = in_sizes[0] / 3;

  // workspace layout (ints): counts[64] | fill[64] | offsets[65] | pad | perm[E]
  int* counts  = (int*)d_ws;
  int* fill    = counts + 64;
  int* offsets = counts + 128;
  int* perm    = counts + 256;

  zero_ws_kernel<<<1, 64, 0, stream>>>(counts, fill);
  int blocks = (E + 255) / 256;
  hist_kernel<<<blocks, 256, 0, stream>>>(trip, counts, E);
  scan_kernel<<<1, 32, 0, stream>>>(counts, offsets);
  scatter_kernel<<<blocks, 256, 0, stream>>>(trip, offsets, fill, perm, E);

  dim3 grid(NRELS, 16);                       // 1024 blocks x 4 waves
  score_kernel<<<grid, WAVES_PER_BLOCK * 32, 0, stream>>>(
      trip, emb, W, offsets, perm, out);
}